// AttentionLayer_11003706213004
// MI455X (gfx1250) — compile-verified
//
#include <hip/hip_runtime.h>

typedef __attribute__((ext_vector_type(2))) float v2f;
typedef __attribute__((ext_vector_type(8))) float v8f;
typedef __attribute__((ext_vector_type(4))) unsigned int v4u;
typedef __attribute__((ext_vector_type(8))) int v8i;
typedef __attribute__((ext_vector_type(4))) int v4i;

#define WMMA_F32(A, B, C) \
  __builtin_amdgcn_wmma_f32_16x16x4_f32(false, (A), false, (B), (short)0, (C), false, false)

// Problem constants (fixed by the reference)
constexpr int Bn = 4, Sn = 16, Hn = 4096, NHn = 32, HDn = 128;
constexpr int PAGEn = 128, CPn = 4096;
constexpr int KBLOCKS = CPn / 16 + 1;  // 256 cached blocks + 1 new-token block

// GEMM tiling: 4 waves per block own a 64-column strip of W; K in 64-chunks.
constexpr int TN = 64;          // W rows (output cols) per block
constexpr int KC = 64;          // K elements per TDM chunk
constexpr int LSTR = KC + 4;    // LDS dword stride (TDM pad: 4 dwords / 64)

// ---------------------------------------------------------------------------
// TDM: issue a 2-D tile load W[row0 .. row0+TN) x [k0 .. k0+KC) -> LDS,
// with hardware padding of 4 dwords after every 64 dwords (bank-conflict-free
// B-fragment reads). ISA 8.3/8.4 descriptor layout; tracked on TENSORcnt.
// This toolchain exposes the 6-arg builtin (clang-23 / therock-10.0 lane).
// ---------------------------------------------------------------------------
__device__ __forceinline__ void tdm_load_wtile(const float* gsrc, void* ldsdst, int K) {
  const unsigned lds = (unsigned)(unsigned long long)ldsdst;   // LDS byte offset
  const unsigned long long ga = (unsigned long long)gsrc;      // global byte addr
  // group0: count=1 | lds_addr | global_addr[56:0] | type=2
  v4u g0 = { 1u, lds, (unsigned)(ga & 0xffffffffu),
             (unsigned)((ga >> 32) & 0x1ffffffu) | (2u << 30) };
  // group1: data_size=4B(2), pad_enable=1, pad_interval=64dw(5), pad_amount=4dw(3)
  //         tensor_dim0=K, tensor_dim1=TN, tile_dim0=KC, tile_dim1=TN,
  //         tensor_dim0_stride=K
  v8i g1 = { (int)((2u << 16) | (1u << 20) | (5u << 22) | (3u << 25)),
             (int)((unsigned)(K & 0xffff) << 16),
             (int)(((unsigned)K >> 16) | ((unsigned)TN << 16)),
             (int)((unsigned)KC << 16),
             (int)TN,
             (int)K,
             0, 0 };
  v4i g2 = { 0, 0, 0, 0 };            // NULL group 2 (2-D tensor)
  v4i g3 = { 0, 0, 0, 0 };            // NULL group 3
  v8i g4 = { 0, 0, 0, 0, 0, 0, 0, 0 };
  __builtin_amdgcn_tensor_load_to_lds(g0, g1, g2, g3, g4, 0);
}

// ---------------------------------------------------------------------------
// Y[64, N] = X[64, K] @ W[N, K]^T + bias
// Weight stream (the bandwidth-critical 265 MB) moved by the Tensor Data
// Mover into double-buffered LDS; waves consume B fragments via ds_load_b64
// and keep 4 M-tile accumulators so every weight byte is moved exactly once.
// ---------------------------------------------------------------------------
__global__ __launch_bounds__(128) void gemm_xwt_kernel(
    const float* __restrict__ X, const float* __restrict__ W,
    const float* __restrict__ bias, float* __restrict__ Y, int N, int K) {
  __shared__ float wtile[2][TN * LSTR];

  const int wave = threadIdx.x >> 5;
  const int lane = threadIdx.x & 31;
  const int col  = lane & 15;   // n within wave tile / m row (A layout)
  const int hf   = lane >> 4;   // lane half selects K-offset 0 vs 2
  const int n0   = blockIdx.x * TN;

  const float* xrow = X + (size_t)col * K + 2 * hf;
  const size_t mK = (size_t)16 * K;
  const float* wlds = &wtile[0][0] + (wave * 16 + col) * LSTR + 2 * hf;

  v8f acc0{}, acc1{}, acc2{}, acc3{};

  const int nchunks = K / KC;
  if (wave == 0) {
    tdm_load_wtile(W + (size_t)n0 * K, &wtile[0][0], K);
    __builtin_amdgcn_s_wait_tensorcnt(0);
  }
  __syncthreads();

  for (int c = 0; c < nchunks; ++c) {
    const int cur = c & 1;
    if (wave == 0 && c + 1 < nchunks)  // async prefetch of next chunk
      tdm_load_wtile(W + (size_t)n0 * K + (c + 1) * KC, &wtile[cur ^ 1][0], K);

    const float* wb = wlds + cur * (TN * LSTR);
    const float* xk = xrow + c * KC;
#pragma unroll
    for (int kk = 0; kk < KC / 4; ++kk) {
      v2f bf = *(const v2f*)(wb + kk * 4);
      v2f a0 = *(const v2f*)(xk + kk * 4);
      v2f a1 = *(const v2f*)(xk + mK + kk * 4);
      v2f a2 = *(const v2f*)(xk + 2 * mK + kk * 4);
      v2f a3 = *(const v2f*)(xk + 3 * mK + kk * 4);
      acc0 = WMMA_F32(a0, bf, acc0);
      acc1 = WMMA_F32(a1, bf, acc1);
      acc2 = WMMA_F32(a2, bf, acc2);
      acc3 = WMMA_F32(a3, bf, acc3);
    }

    if (wave == 0 && c + 1 < nchunks) __builtin_amdgcn_s_wait_tensorcnt(0);
    __syncthreads();
  }

  const int n = n0 + wave * 16 + col;
  const float bv = bias[n];
  float* y = Y + (size_t)(8 * hf) * N + n;
#pragma unroll
  for (int r = 0; r < 8; ++r) {
    y[(size_t)r * N]        = acc0[r] + bv;
    y[(size_t)(16 + r) * N] = acc1[r] + bv;
    y[(size_t)(32 + r) * N] = acc2[r] + bv;
    y[(size_t)(48 + r) * N] = acc3[r] + bv;
  }
}

// ---------------------------------------------------------------------------
// Flash attention over paged KV cache. One wave per (batch, head); 8 waves
// per block. Q fragments live in registers for the whole sweep; the P block
// is re-laid out C-layout -> A-layout through a private LDS tile.
// ---------------------------------------------------------------------------
__global__ __launch_bounds__(256) void attn_kernel(
    const float* __restrict__ qkv, const float* __restrict__ k_pages,
    const float* __restrict__ v_pages, const int* __restrict__ page_table,
    float* __restrict__ attn_out) {
  __shared__ float psm[8 * 256];  // 16x16 P tile per wave

  const int wave = threadIdx.x >> 5;
  const int lane = threadIdx.x & 31;
  const int col  = lane & 15;
  const int hf   = lane >> 4;
  const int w    = blockIdx.x * 8 + wave;  // 0..127
  const int b    = w >> 5;                 // batch
  const int h    = w & 31;                 // head
  float* P = psm + wave * 256;

  // --- Q fragments (16 queries x 128 dims, pre-scaled by 1/sqrt(HD)) ---
  const float scale = 0.08838834764831845f;  // 1/sqrt(128)
  const float* qrow = qkv + (size_t)(b * Sn + col) * (3 * Hn) + h * HDn + 2 * hf;
  v2f qf[32];
#pragma unroll
  for (int c = 0; c < 32; ++c) {
    v2f t = *(const v2f*)(qrow + 4 * c);
    qf[c].x = t.x * scale;
    qf[c].y = t.y * scale;
  }

  v8f o[8];
#pragma unroll
  for (int nt = 0; nt < 8; ++nt) o[nt] = v8f{};
  float m[8], l[8];
#pragma unroll
  for (int r = 0; r < 8; ++r) { m[r] = -3.402823466e38f; l[r] = 0.f; }

  for (int t = 0; t < KBLOCKS; ++t) {
    const float* kbase;
    const float* vbase;
    size_t rstride;
    if (t < CPn / 16) {
      const int page = page_table[t >> 3];           // 16-key block lies in one page
      const int off  = (t & 7) * 16;                 // offset within page
      const size_t base = ((size_t)(page * PAGEn + off) * NHn + h) * HDn;
      kbase = k_pages + base;
      vbase = v_pages + base;
      rstride = (size_t)NHn * HDn;                   // 4096 floats between positions
    } else {
      const size_t base = (size_t)(b * Sn) * (3 * Hn) + Hn + h * HDn;
      kbase = qkv + base;            // new K rows
      vbase = qkv + base + Hn;       // new V rows
      rstride = 3 * Hn;              // 12288
    }

    // --- scores S = Q @ K^T for this 16-key block (C layout: q rows, key cols)
    const float* krow = kbase + (size_t)col * rstride + 2 * hf;
    v8f s{};
#pragma unroll
    for (int c = 0; c < 32; ++c) {
      v2f bf = *(const v2f*)(krow + 4 * c);
      s = WMMA_F32(qf[c], bf, s);
    }

    // --- online softmax; rows q = r + 8*hf live entirely within a lane-half
#pragma unroll
    for (int r = 0; r < 8; ++r) {
      float v = s[r];
      v = fmaxf(v, __shfl_xor(v, 1, 32));
      v = fmaxf(v, __shfl_xor(v, 2, 32));
      v = fmaxf(v, __shfl_xor(v, 4, 32));
      v = fmaxf(v, __shfl_xor(v, 8, 32));
      const float mn = fmaxf(m[r], v);
      const float sc = __expf(m[r] - mn);
      const float p  = __expf(s[r] - mn);
      float rs = p;
      rs += __shfl_xor(rs, 1, 32);
      rs += __shfl_xor(rs, 2, 32);
      rs += __shfl_xor(rs, 4, 32);
      rs += __shfl_xor(rs, 8, 32);
      l[r] = l[r] * sc + rs;
      m[r] = mn;
#pragma unroll
      for (int nt = 0; nt < 8; ++nt) o[nt][r] *= sc;
      P[(r + 8 * hf) * 16 + col] = p;  // C-layout -> row-major tile in LDS
    }

    // --- A fragments of P (same-wave LDS ops are in-order; store->load safe)
    v2f af[4];
#pragma unroll
    for (int kc = 0; kc < 4; ++kc)
      af[kc] = *(const v2f*)(P + col * 16 + kc * 4 + 2 * hf);

    // --- O += P @ V
#pragma unroll
    for (int kc = 0; kc < 4; ++kc) {
      const float* vr0 = vbase + (size_t)(4 * kc + 2 * hf) * rstride;
      const float* vr1 = vr0 + rstride;
#pragma unroll
      for (int nt = 0; nt < 8; ++nt) {
        v2f bf;
        bf.x = vr0[nt * 16 + col];
        bf.y = vr1[nt * 16 + col];
        o[nt] = WMMA_F32(af[kc], bf, o[nt]);
      }
    }
  }

  // --- epilogue: O /= l, write [b, q, h, d] (== [b, s, 4096] row-major)
#pragma unroll
  for (int r = 0; r < 8; ++r) {
    const float inv = 1.0f / l[r];
    const int q = r + 8 * hf;
    float* orow = attn_out + (size_t)(b * Sn + q) * Hn + h * HDn + col;
#pragma unroll
    for (int nt = 0; nt < 8; ++nt) orow[nt * 16] = o[nt][r] * inv;
  }
}

// ---------------------------------------------------------------------------
extern "C" void kernel_launch(void* const* d_in, const int* in_sizes, int n_in,
                              void* d_out, int out_size, void* d_ws, size_t ws_size,
                              hipStream_t stream) {
  const float* x     = (const float*)d_in[0];
  const float* Wqkv  = (const float*)d_in[1];
  const float* bqkv  = (const float*)d_in[2];
  const float* Wproj = (const float*)d_in[3];
  const float* bproj = (const float*)d_in[4];
  const float* kp    = (const float*)d_in[5];
  const float* vp    = (const float*)d_in[6];
  const int*   pt    = (const int*)d_in[7];
  float* out = (float*)d_out;

  float* qkv  = (float*)d_ws;                        // [64, 12288]
  float* attn = qkv + (size_t)Bn * Sn * 3 * Hn;      // [64, 4096]

  // 1) QKV projection: [64,4096] @ [12288,4096]^T + b   (TDM-fed WMMA)
  gemm_xwt_kernel<<<dim3(3 * Hn / TN), 128, 0, stream>>>(x, Wqkv, bqkv, qkv, 3 * Hn, Hn);

  // 2) Paged flash attention: 128 (b,h) waves
  attn_kernel<<<dim3(Bn * NHn / 8), 256, 0, stream>>>(qkv, kp, vp, pt, attn);

  // 3) Output projection: [64,4096] @ [4096,4096]^T + b (TDM-fed WMMA)
  gemm_xwt_kernel<<<dim3(Hn / TN), 128, 0, stream>>>(attn, Wproj, bproj, out, Hn, Hn);
}